// NNODEModel_20899310862861
// MI455X (gfx1250) — compile-verified
//
#include <hip/hip_runtime.h>
#include <math.h>

typedef __attribute__((ext_vector_type(2))) float v2f;
typedef __attribute__((ext_vector_type(8))) float v8f;

#define WAVES 8
#define THREADS (WAVES * 32)
#define DDIM 32
#define RS 33                                 // padded x-tile row stride (dwords)
#define MT 4                                  // 16-row M-tiles per wave
#define ROWS_PER_WAVE (MT * 16)               // 64
#define ROWS_PER_WG (WAVES * ROWS_PER_WAVE)   // 512

#define NPAIR 528                             // upper-triangular (i<=j) pairs
#define KSYM 576                              // row-padded symmetric K (mult of 4)
#define KROWS (KSYM + DDIM)                   // + 32 linear rows: 608
#define NCHUNK (KROWS / 4)                    // 152 K-chunks
#define CHUNK_DW 128                          // dwords per packed B chunk block

// LDS layout (floats)
#define OFF_B 0                               // packed B: NCHUNK * CHUNK_DW
#define OFF_X (NCHUNK * CHUNK_DW)             // per-wave x tiles (19456)
#define LDS_FLOATS (OFF_X + WAVES * ROWS_PER_WAVE * RS + 8)  // ~142 KB

__device__ __forceinline__ v8f wmma4(v2f a, v2f b, v8f c) {
  // D = A(16x4 f32) * B(4x16 f32) + C(16x16 f32)
  return __builtin_amdgcn_wmma_f32_16x16x4_f32(false, a, false, b, (short)0, c,
                                               false, false);
}

__device__ __forceinline__ int tri(int i) { return i * (65 - i) / 2; }  // pairs before row i
__device__ __forceinline__ int rowstart(int i) {                         // padded K start of row i
  int a = i >> 2, b = i & 3;
  return 32 * i - 8 * a * (a - 1) - 4 * a * b;
}
// packed-B address (dwords) of element B[kpos][o]
__device__ __forceinline__ int bp_addr(int kpos, int o) {
  int c = kpos >> 2, r = kpos & 3, hh = r >> 1, s = r & 1;
  return c * CHUNK_DW + hh * 64 + 2 * ((o + 16 * hh) & 31) + s;
}

__global__ __launch_bounds__(THREADS) void nnode_bilinear_kernel(
    const float* __restrict__ x, const float* __restrict__ Wlin,
    const float* __restrict__ Wnl, float* __restrict__ out) {
  extern __shared__ float lds[];
  const int tid  = threadIdx.x;
  const int wave = tid >> 5;
  const int lane = tid & 31;
  const int m    = lane & 15;   // N column / M row within a tile
  const int h    = lane >> 4;   // wave half
  const int h2   = h * 2;

  // ---- Zero packed-B region (covers pad slots) + x tail pad
  for (int idx = tid; idx < NCHUNK * CHUNK_DW; idx += THREADS) lds[OFF_B + idx] = 0.0f;
  if (tid < 8) lds[OFF_X + WAVES * ROWS_PER_WAVE * RS + tid] = 0.0f;

  // ---- Stage this wave's 64 x-rows
  const int rowBase = blockIdx.x * ROWS_PER_WG + wave * ROWS_PER_WAVE;
  float* xsw = &lds[OFF_X + wave * ROWS_PER_WAVE * RS];
  {
    const float4* x4 = (const float4*)(x + rowBase * DDIM);
    #pragma unroll
    for (int it = 0; it < (ROWS_PER_WAVE * DDIM) / (32 * 4); ++it) {
      int flat4 = it * 32 + lane;
      int r = flat4 >> 3;
      int c = (flat4 & 7) * 4;
      float4 v = x4[flat4];
      xsw[r * RS + c + 0] = v.x;
      xsw[r * RS + c + 1] = v.y;
      xsw[r * RS + c + 2] = v.z;
      xsw[r * RS + c + 3] = v.w;
    }
  }
  __syncthreads();   // zeroed B region visible before fills

  // ---- Symmetric B fill: pair (i<=j) -> chunk-packed, half-rotated layout
  {
    #pragma unroll 2
    for (int it = 0; it < (NPAIR * DDIM) / THREADS; ++it) {   // 66 iters
      int idx = it * THREADS + tid;      // p*32 + o
      int o = idx & 31;
      int p = idx >> 5;                  // 0..527
      int i = (int)((65.0f - sqrtf(4225.0f - 8.0f * (float)p)) * 0.5f);
      while (i > 0 && tri(i) > p) --i;
      while (tri(i + 1) <= p) ++i;
      int j = i + (p - tri(i));
      float v = Wnl[o * 1024 + i * 32 + j];
      if (j > i) v += Wnl[o * 1024 + j * 32 + i];
      lds[OFF_B + bp_addr(rowstart(i) + (j - i), o)] = v;
    }
  }
  // ---- Linear rows: B[KSYM + k][o] = Wlin[k][o]
  {
    #pragma unroll
    for (int it = 0; it < (DDIM * DDIM) / THREADS; ++it) {
      int idx = it * THREADS + tid;      // k*32 + o
      lds[OFF_B + bp_addr(KSYM + (idx >> 5), idx & 31)] = Wlin[idx];
    }
  }
  __syncthreads();

  // Per-lane packed-B pointers: one ds_load_b64 per WMMA B operand.
  const float* p0 = &lds[OFF_B + h * 64 + 2 * lane];          // N-tile 0
  const float* p1 = &lds[OFF_B + h * 64 + 2 * (lane ^ 16)];   // N-tile 1

  v8f acc[MT][2] = {};  // [M-tile][N-tile]

  // ---- Symmetric bilinear term: rows grouped by equal chunk count
  int cbase = 0;                        // dword offset of current row's chunks
  #pragma unroll
  for (int g = 0; g < 8; ++g) {
    const int nch = 8 - g;              // chunks per row in this group
    #pragma unroll 1                    // keep rolled: bounds register pressure
    for (int rr = 0; rr < 4; ++rr) {
      const int i = g * 4 + rr;
      float xi[MT];
      #pragma unroll
      for (int t = 0; t < MT; ++t) xi[t] = xsw[(t * 16 + m) * RS + i];
      #pragma unroll
      for (int cc = 0; cc < nch; ++cc) {
        v2f Bn0 = *(const v2f*)(p0 + cbase + cc * CHUNK_DW);
        v2f Bn1 = *(const v2f*)(p1 + cbase + cc * CHUNK_DW);
        #pragma unroll
        for (int t = 0; t < MT; ++t) {
          const float* ap = &xsw[(t * 16 + m) * RS + i + 4 * cc + h2];
          v2f A = {xi[t] * ap[0], xi[t] * ap[1]};   // pad slots hit zero B
          acc[t][0] = wmma4(A, Bn0, acc[t][0]);
          acc[t][1] = wmma4(A, Bn1, acc[t][1]);
        }
      }
      cbase += nch * CHUNK_DW;
    }
  }

  // ---- Linear tail: A = x itself, chunks [KSYM/4, KSYM/4 + 8)
  #pragma unroll
  for (int cc = 0; cc < 8; ++cc) {
    v2f Bn0 = *(const v2f*)(p0 + (KSYM / 4 + cc) * CHUNK_DW);
    v2f Bn1 = *(const v2f*)(p1 + (KSYM / 4 + cc) * CHUNK_DW);
    #pragma unroll
    for (int t = 0; t < MT; ++t) {
      const float* ap = &xsw[(t * 16 + m) * RS + 4 * cc + h2];
      v2f A = {ap[0], ap[1]};
      acc[t][0] = wmma4(A, Bn0, acc[t][0]);
      acc[t][1] = wmma4(A, Bn1, acc[t][1]);
    }
  }

  // ---- Store D: vgpr r holds M = r (lanes 0-15) / r+8 (lanes 16-31)
  #pragma unroll
  for (int t = 0; t < MT; ++t) {
    #pragma unroll
    for (int r = 0; r < 8; ++r) {
      int row = rowBase + t * 16 + h * 8 + r;
      float* op = out + row * DDIM + m;
      op[0]  = acc[t][0][r];
      op[16] = acc[t][1][r];
    }
  }
}

extern "C" void kernel_launch(void* const* d_in, const int* in_sizes, int n_in,
                              void* d_out, int out_size, void* d_ws, size_t ws_size,
                              hipStream_t stream) {
  const float* x    = (const float*)d_in[0];
  const float* Wlin = (const float*)d_in[1];
  const float* Wnl  = (const float*)d_in[2];
  float* out = (float*)d_out;

  int Btot   = in_sizes[0] / DDIM;        // 524288
  int blocks = Btot / ROWS_PER_WG;        // 1024
  size_t shmem = (size_t)LDS_FLOATS * sizeof(float);  // ~142 KB dynamic LDS

  nnode_bilinear_kernel<<<blocks, THREADS, shmem, stream>>>(x, Wlin, Wnl, out);
}